// Tem_AttentionLayer_29583734735559
// MI455X (gfx1250) — compile-verified
//
#include <hip/hip_runtime.h>
#include <hip/hip_bf16.h>
#include <math.h>

// MI455X / gfx1250, wave32. WMMA f32 16x16x4 used for all matmuls (fp32-exact).
typedef __attribute__((ext_vector_type(2))) float v2f;
typedef __attribute__((ext_vector_type(8))) float v8f;

#define SEQ    384    // B*s2 = GRU batch
#define LSTEPS 325    // s1 = GRU length
#define DIM    128
#define G3D    384    // 3*DIM gates
#define HPAD   132    // padded row stride (4-word skew -> conflict-free WMMA A loads)
#define SPAD   388    // padded stride for 384-col gate buffer

struct GruArgs {
    const float* x[3];
    const float* Wih[3];
    const float* Whh[3];
    const float* bih[3];
    const float* bhh[3];
    float*       out[3];   // [LSTEPS][SEQ][DIM]
};

// One WG = 16 sequences of one GRU, persistent over all 325 steps.
// 8 waves; wave w owns r/z tiles {2w, 2w+1} (cols 0..255) and n tile {16+w} (cols
// 256..383). r/z products of x*Wih^T and h*Whh^T are chained into ONE accumulator via
// WMMA's C operand (the gates only need their sum); only the n tile keeps P/Q split.
// Whh is register-resident (96 v2f); Wih streams from the 192MB L2 each step.
__global__ __launch_bounds__(256, 1) void gru_persistent_kernel(GruArgs a) {
    const int which = blockIdx.x / 24;          // 0=q 1=k 2=v
    const int r0    = (blockIdx.x % 24) * 16;   // first sequence of this M-tile

    const float* __restrict__ x   = a.x[which];
    const float* __restrict__ Wih = a.Wih[which];
    const float* __restrict__ Whh = a.Whh[which];
    const float* __restrict__ bih = a.bih[which];
    const float* __restrict__ bhh = a.bhh[which];
    float* __restrict__ out       = a.out[which];

    __shared__ float h_lds[16 * HPAD];   // hidden state (fp32, persists across steps)
    __shared__ float x_lds[16 * HPAD];   // x_t tile
    __shared__ float S_lds[16 * SPAD];   // r/z: ih+hh summed; n: ih part
    __shared__ float Qn_lds[16 * HPAD];  // n: hh part (needs the r-gate multiplier)
    __shared__ float bih_l[G3D], bhh_l[G3D];

    const int tid  = threadIdx.x;
    const int lane = tid & 31;
    const int w    = tid >> 5;
    const int half = lane >> 4;
    const int nl   = lane & 15;

    // This wave's three column tiles: two r/z + one n.
    const int tileg[3] = { 2 * w, 2 * w + 1, 16 + w };

    for (int i = tid; i < G3D; i += 256) { bih_l[i] = bih[i]; bhh_l[i] = bhh[i]; }
    for (int i = tid; i < 16 * HPAD; i += 256) h_lds[i] = 0.0f;

    // ---- one-time: preload this wave's Whh B-operands into registers ----
    // B[k][n] = W[g][k] with g = tile*16 + nl; lane supplies K = 4*kk4 + 2*half + {0,1}.
    v2f whh_r[96];
    #pragma unroll
    for (int kk4 = 0; kk4 < 32; ++kk4) {
        #pragma unroll
        for (int j = 0; j < 3; ++j) {
            const int g  = tileg[j] * 16 + nl;
            const int ko = kk4 * 4 + 2 * half;
            whh_r[kk4 * 3 + j] = *(const v2f*)&Whh[(size_t)g * DIM + ko];
        }
    }
    // Per-lane Wih base pointers (3 column tiles); loads use immediate offsets.
    const float* wih_lane[3];
    #pragma unroll
    for (int j = 0; j < 3; ++j)
        wih_lane[j] = Wih + (size_t)(tileg[j] * 16 + nl) * DIM + 2 * half;

    // ---- software pipeline: x(0) into registers ----
    float xreg[8];
    #pragma unroll
    for (int i = 0; i < 8; ++i) {
        const int idx = i * 256 + tid;
        const int m = idx >> 7, d = idx & 127;
        xreg[i] = x[((size_t)(r0 + m) * LSTEPS + 0) * DIM + d];
    }
    __syncthreads();

    for (int n = 0; n < LSTEPS; ++n) {
        // ---- commit prefetched x_t to LDS ----
        #pragma unroll
        for (int i = 0; i < 8; ++i) {
            const int idx = i * 256 + tid;
            const int m = idx >> 7, d = idx & 127;
            x_lds[m * HPAD + d] = xreg[i];
        }
        __syncthreads();

        // ---- WMMA: r/z tiles accumulate h*Whh^T and x*Wih^T into one C;
        //      n tile keeps the two products split (gate math needs r*hn) ----
        v8f accRZ0 = (v8f){}, accRZ1 = (v8f){}, accPn = (v8f){}, accQn = (v8f){};

        #pragma unroll
        for (int kk4 = 0; kk4 < 32; ++kk4) {
            // Cap VMEM hoisting at the midpoint (ALU/WMMA/DS may cross freely).
            if (kk4 == 16) __builtin_amdgcn_sched_barrier(0x78F);
            const int ko = kk4 * 4 + 2 * half;
            v2f ax = *(const v2f*)&x_lds[nl * HPAD + ko]; // A: row m=nl, K=ko..ko+1
            v2f ah = *(const v2f*)&h_lds[nl * HPAD + ko];
            v2f bi0 = *(const v2f*)&wih_lane[0][kk4 * 4];
            v2f bi1 = *(const v2f*)&wih_lane[1][kk4 * 4];
            v2f bi2 = *(const v2f*)&wih_lane[2][kk4 * 4];
            accRZ0 = __builtin_amdgcn_wmma_f32_16x16x4_f32(
                false, ah, false, whh_r[kk4 * 3 + 0], (short)0, accRZ0, false, false);
            accRZ0 = __builtin_amdgcn_wmma_f32_16x16x4_f32(
                false, ax, false, bi0, (short)0, accRZ0, false, false);
            accRZ1 = __builtin_amdgcn_wmma_f32_16x16x4_f32(
                false, ah, false, whh_r[kk4 * 3 + 1], (short)0, accRZ1, false, false);
            accRZ1 = __builtin_amdgcn_wmma_f32_16x16x4_f32(
                false, ax, false, bi1, (short)0, accRZ1, false, false);
            accQn = __builtin_amdgcn_wmma_f32_16x16x4_f32(
                false, ah, false, whh_r[kk4 * 3 + 2], (short)0, accQn, false, false);
            accPn = __builtin_amdgcn_wmma_f32_16x16x4_f32(
                false, ax, false, bi2, (short)0, accPn, false, false);
        }
        __syncthreads();   // matmul reads of h_lds complete before gate phase rewrites h

        // ---- issue next step's x loads (complete during spill + gate phases) ----
        const int np = (n + 1 < LSTEPS) ? n + 1 : n;
        #pragma unroll
        for (int i = 0; i < 8; ++i) {
            const int idx = i * 256 + tid;
            const int m = idx >> 7, d = idx & 127;
            xreg[i] = x[((size_t)(r0 + m) * LSTEPS + np) * DIM + d];
        }

        // ---- spill gate tiles to LDS ----
        {
            const int gb0 = tileg[0] * 16 + nl;
            const int gb1 = tileg[1] * 16 + nl;
            const int gbn = tileg[2] * 16 + nl;            // 256 + w*16 + nl
            #pragma unroll
            for (int v = 0; v < 8; ++v) {
                const int m = v + 8 * half;
                S_lds[m * SPAD + gb0]          = accRZ0[v];    // xr+hr / xz+hz
                S_lds[m * SPAD + gb1]          = accRZ1[v];
                S_lds[m * SPAD + gbn]          = accPn[v];     // xn
                Qn_lds[m * HPAD + (gbn - 256)] = accQn[v];     // hn
            }
        }
        __syncthreads();

        // ---- gate math: h' = (1-z)*tanh(xn + r*hn) + z*h ----
        #pragma unroll
        for (int i = 0; i < 8; ++i) {
            const int idx = i * 256 + tid;
            const int m = idx >> 7, d = idx & 127;
            float rg = S_lds[m * SPAD + d]       + bih_l[d]       + bhh_l[d];
            float zg = S_lds[m * SPAD + 128 + d] + bih_l[128 + d] + bhh_l[128 + d];
            float xn = S_lds[m * SPAD + 256 + d] + bih_l[256 + d];
            float hn = Qn_lds[m * HPAD + d]      + bhh_l[256 + d];
            float r  = 1.0f / (1.0f + __expf(-rg));
            float z  = 1.0f / (1.0f + __expf(-zg));
            float nn = tanhf(xn + r * hn);
            float hv = (1.0f - z) * nn + z * h_lds[m * HPAD + d];
            h_lds[m * HPAD + d] = hv;                      // (m,d) exclusive to this thread
            out[((size_t)n * SEQ + r0 + m) * DIM + d] = hv; // [L][SEQ][D] for attention
        }
        __syncthreads();
    }
}

// One WG per (b,n): 8-head 12x12 attention (VALU) + 128x128 output GEMM (WMMA f32).
__global__ __launch_bounds__(256, 1) void attn_out_kernel(
    const float* __restrict__ qo, const float* __restrict__ kin,
    const float* __restrict__ vo, const float* __restrict__ rel_pos,
    const float* __restrict__ Wout, const float* __restrict__ bout,
    float* __restrict__ dst) {
    const int b = blockIdx.x / LSTEPS;
    const int n = blockIdx.x % LSTEPS;

    __shared__ float Qs[12 * DIM], Ks[12 * DIM], Vs[12 * DIM];
    __shared__ float att[16 * HPAD];   // heads re-concatenated, padded to M=16

    const int tid = threadIdx.x;
    const size_t base = ((size_t)n * SEQ + b * 12) * DIM;   // rows b*12..b*12+11 contiguous
    for (int i = tid; i < 12 * DIM; i += 256) {
        Qs[i] = qo[base + i]; Ks[i] = kin[base + i]; Vs[i] = vo[base + i];
    }
    for (int i = tid; i < 16 * HPAD; i += 256) att[i] = 0.0f;
    __syncthreads();

    if (tid < 96) {  // one thread per (head h, row t)
        const int h = tid & 7, t = tid >> 3;
        const int h2 = (5 * b + n) & 7;   // scrambled rel_pos head (derived from reshape)
        const float* bias = rel_pos + h2 * 144 + t * 12;
        float sc[12], mx = -1e30f;
        #pragma unroll
        for (int j = 0; j < 12; ++j) {
            float s = 0.0f;
            #pragma unroll
            for (int e = 0; e < 16; ++e)
                s += Qs[t * DIM + h * 16 + e] * Ks[j * DIM + h * 16 + e];
            s = s * 0.25f + bias[j];       // 1/sqrt(16)
            sc[j] = s; mx = fmaxf(mx, s);
        }
        float sum = 0.0f;
        #pragma unroll
        for (int j = 0; j < 12; ++j) { sc[j] = __expf(sc[j] - mx); sum += sc[j]; }
        const float inv = 1.0f / sum;
        #pragma unroll
        for (int e = 0; e < 16; ++e) {
            float o = 0.0f;
            #pragma unroll
            for (int j = 0; j < 12; ++j) o += sc[j] * Vs[j * DIM + h * 16 + e];
            att[t * HPAD + h * 16 + e] = o * inv;
        }
    }
    __syncthreads();

    // ---- out = att[12,128] * Wout^T + bout via WMMA f32; wave w owns 16 cols ----
    const int lane = tid & 31, w = tid >> 5;
    const int half = lane >> 4, nl = lane & 15;
    const int col  = w * 16 + nl;
    v8f acc = (v8f){};
    #pragma unroll 8
    for (int kk = 0; kk < DIM; kk += 4) {
        const int ko = kk + 2 * half;
        v2f amat = *(const v2f*)&att[nl * HPAD + ko];
        v2f bmat = *(const v2f*)&Wout[(size_t)col * DIM + ko];
        acc = __builtin_amdgcn_wmma_f32_16x16x4_f32(
            false, amat, false, bmat, (short)0, acc, false, false);
    }
    const float bo = bout[col];
    const size_t ob = ((size_t)b * LSTEPS + n) * 12;
    #pragma unroll
    for (int v = 0; v < 8; ++v) {
        const int m = v + 8 * half;
        if (m < 12) dst[(ob + m) * DIM + col] = acc[v] + bo;
    }
}

extern "C" void kernel_launch(void* const* d_in, const int* in_sizes, int n_in,
                              void* d_out, int out_size, void* d_ws, size_t ws_size,
                              hipStream_t stream) {
    // setup_inputs order: query,key,value, {Wih,Whh,bih,bhh} x (q,k,v), rel_pos, Wout, bout
    const float* query = (const float*)d_in[0];
    const float* key   = (const float*)d_in[1];
    const float* value = (const float*)d_in[2];
    const float* rel   = (const float*)d_in[15];
    const float* Wout  = (const float*)d_in[16];
    const float* bout  = (const float*)d_in[17];

    float* ws = (float*)d_ws;
    const size_t one = (size_t)LSTEPS * SEQ * DIM;   // 15,974,400 floats
    float* qo = ws;
    float* ko = ws + one;
    float* vo = ws + 2 * one;

    GruArgs ga;
    const float* xs[3] = {query, key, value};
    float* os[3] = {qo, ko, vo};
    for (int g = 0; g < 3; ++g) {
        ga.x[g]   = xs[g];
        ga.Wih[g] = (const float*)d_in[3 + 4 * g + 0];
        ga.Whh[g] = (const float*)d_in[3 + 4 * g + 1];
        ga.bih[g] = (const float*)d_in[3 + 4 * g + 2];
        ga.bhh[g] = (const float*)d_in[3 + 4 * g + 3];
        ga.out[g] = os[g];
    }

    gru_persistent_kernel<<<72, 256, 0, stream>>>(ga);
    attn_out_kernel<<<32 * LSTEPS, 256, 0, stream>>>(qo, ko, vo, rel, Wout, bout,
                                                     (float*)d_out);
}